// Model_17686675325086
// MI455X (gfx1250) — compile-verified
//
#include <hip/hip_runtime.h>

#define D_MODEL 128
#define C_FEAT  16
#define LN_EPS  1e-5f

typedef __attribute__((ext_vector_type(2))) float v2f;
typedef __attribute__((ext_vector_type(8))) float v8f;

// Branchless erf, Abramowitz-Stegun 7.1.26 (max abs err ~1.5e-7).
// Avoids libm erff's exec-mask divergent two-region implementation.
__device__ __forceinline__ float erf_fast(float x) {
    const float ax = fabsf(x);
    const float t  = __builtin_amdgcn_rcpf(fmaf(0.3275911f, ax, 1.0f));
    float p = fmaf(1.061405429f, t, -1.453152027f);
    p = fmaf(p, t, 1.421413741f);
    p = fmaf(p, t, -0.284496736f);
    p = fmaf(p, t, 0.254829592f);
    const float y = 1.0f - p * t * __expf(-ax * ax);
    return copysignf(y, x);
}

__device__ __forceinline__ float gelu_exact(float x) {
    // torch F.gelu exact: 0.5*x*(1+erf(x/sqrt(2)))
    return 0.5f * x * (1.0f + erf_fast(x * 0.70710678118654752f));
}

// Fused Linear(16->128) + GELU + LayerNorm over D=128.
// One wave per 16-row tile. Four V_WMMA_F32_16X16X4_F32 per 16x16 N-tile,
// 8 N-tiles cover D=128. LayerNorm done in-register via half-wave shuffles.
__global__ __launch_bounds__(128) void seg_encoder_wmma(
    const float* __restrict__ X,     // [R, 16]
    const float* __restrict__ W,     // [16, 128]
    const float* __restrict__ bias,  // [128]
    const float* __restrict__ lng,   // [128]
    const float* __restrict__ lnb,   // [128]
    float* __restrict__ H0,          // [R, 128]
    int R)
{
    const int lane = threadIdx.x & 31;
    const int wave = threadIdx.x >> 5;
    const int tile = blockIdx.x * (blockDim.x >> 5) + wave;
    const int row0 = tile << 4;
    if (row0 >= R) return;                 // whole-wave exit; EXEC stays full

    const int m  = lane & 15;              // row within tile / column within N-tile
    const int hi = lane >> 4;              // half-wave select

    // ---- A tile: 16 rows x 16 K, packed for 4 K-steps of 16x16x4 ----
    int arowi = row0 + m; if (arowi >= R) arowi = R - 1;   // clamp (tail safety)
    const float* arow = X + (size_t)arowi * C_FEAT;
    v2f a[4];
#pragma unroll
    for (int k = 0; k < 4; ++k)
        a[k] = *(const v2f*)(arow + 4 * k + 2 * hi);       // {K=4k+2hi, K=4k+2hi+1}

    float h[8][8];          // h[t][r]: tile t (N = 16t + m), accumulator reg r
    float gv[8], bv[8];

#pragma unroll
    for (int t = 0; t < 8; ++t) {
        const int n = (t << 4) + m;
        v8f acc = {0.f, 0.f, 0.f, 0.f, 0.f, 0.f, 0.f, 0.f};
#pragma unroll
        for (int k = 0; k < 4; ++k) {
            const int kk = 4 * k + 2 * hi;
            v2f b;
            b.x = W[kk * D_MODEL + n];
            b.y = W[(kk + 1) * D_MODEL + n];
            acc = __builtin_amdgcn_wmma_f32_16x16x4_f32(
                false, a[k], false, b, (short)0, acc, false, false);
        }
        const float bn = bias[n];
        gv[t] = lng[n];
        bv[t] = lnb[n];
#pragma unroll
        for (int r = 0; r < 8; ++r)
            h[t][r] = gelu_exact(acc[r] + bn);
    }

    // ---- LayerNorm across D=128 for row M = r + 8*hi ----
#pragma unroll
    for (int r = 0; r < 8; ++r) {
        float s = 0.f, q = 0.f;
#pragma unroll
        for (int t = 0; t < 8; ++t) { s += h[t][r]; q += h[t][r] * h[t][r]; }
        // reduce across the 16 lanes holding this row (masks 1..8 stay in half-wave)
#pragma unroll
        for (int mask = 1; mask <= 8; mask <<= 1) {
            s += __shfl_xor(s, mask, 32);
            q += __shfl_xor(q, mask, 32);
        }
        const float mu   = s * (1.f / 128.f);
        const float var  = q * (1.f / 128.f) - mu * mu;
        const float rstd = rsqrtf(var + LN_EPS);
        const int rowM = row0 + r + 8 * hi;
        if (rowM < R) {
            float* orow = H0 + (size_t)rowM * D_MODEL + m;
#pragma unroll
            for (int t = 0; t < 8; ++t)
                orow[t << 4] = (h[t][r] - mu) * rstd * gv[t] + bv[t];
        }
    }
}

// Compact xy table: [B*N, 2] from static[..., :2] -> 8B-granule L2-resident gathers.
__global__ void xy_compact(const float* __restrict__ X, float2* __restrict__ xy, int BN) {
    int i = blockIdx.x * blockDim.x + threadIdx.x;
    if (i < BN) {
        float2 p;
        p.x = X[(size_t)i * C_FEAT + 0];
        p.y = X[(size_t)i * C_FEAT + 1];
        xy[i] = p;
    }
}

__global__ void edge_det_compact(const int* __restrict__ ei,       // [2, E]
                                 const float2* __restrict__ xy,    // [B*N]
                                 float* __restrict__ det,          // [B*E]
                                 int E, int N) {
    int e = blockIdx.x * blockDim.x + threadIdx.x;
    int b = blockIdx.y;
    if (e >= E) return;
    const int off = b * N;
    const int s = ei[e] + off;
    const int d = ei[E + e] + off;
    const float2 ps = xy[s];
    const float2 pd = xy[d];
    const float dx = ps.x - pd.x, dy = ps.y - pd.y;
    det[(size_t)b * E + e] = -0.5f * (dx * dx + dy * dy);   // SIGMA = 1
}

// Fallback if workspace is too small: gather straight from static (stride 16).
__global__ void edge_det_direct(const int* __restrict__ ei,
                                const float* __restrict__ X,
                                float* __restrict__ det,
                                int E, int N) {
    int e = blockIdx.x * blockDim.x + threadIdx.x;
    int b = blockIdx.y;
    if (e >= E) return;
    const int off = b * N;
    const size_t s = (size_t)(ei[e] + off) * C_FEAT;
    const size_t d = (size_t)(ei[E + e] + off) * C_FEAT;
    const float dx = X[s] - X[d];
    const float dy = X[s + 1] - X[d + 1];
    det[(size_t)b * E + e] = -0.5f * (dx * dx + dy * dy);
}

extern "C" void kernel_launch(void* const* d_in, const int* in_sizes, int n_in,
                              void* d_out, int out_size, void* d_ws, size_t ws_size,
                              hipStream_t stream) {
    const float* statics = (const float*)d_in[0];   // [B, N, 16]
    const int*   ei      = (const int*)d_in[1];     // [2, E]
    const float* fc_w    = (const float*)d_in[2];   // [16, 128]
    const float* fc_b    = (const float*)d_in[3];   // [128]
    const float* ln_g    = (const float*)d_in[4];   // [128]
    const float* ln_b    = (const float*)d_in[5];   // [128]
    float* out = (float*)d_out;

    const int BN = in_sizes[0] / C_FEAT;            // B*N = 100000
    const int E  = in_sizes[1] / 2;                 // 1,600,000
    const int B  = 2;
    const int N  = BN / B;

    float* H0  = out;                                // [B*N, 128]
    float* det = out + (size_t)BN * D_MODEL;         // [B*E]

    // ---- Fused encoder (WMMA GEMM + GELU + LayerNorm) ----
    {
        const int tiles = (BN + 15) >> 4;            // 6250
        const int wavesPerBlock = 4;                 // 128 threads = 4 wave32
        dim3 block(32 * wavesPerBlock);
        dim3 grid((tiles + wavesPerBlock - 1) / wavesPerBlock);
        seg_encoder_wmma<<<grid, block, 0, stream>>>(
            statics, fc_w, fc_b, ln_g, ln_b, H0, BN);
    }

    // ---- Edge deterrence ----
    const size_t xybytes = (size_t)BN * sizeof(float2);
    dim3 egrid((E + 255) / 256, B);
    if (ws_size >= xybytes) {
        float2* xy = (float2*)d_ws;
        xy_compact<<<(BN + 255) / 256, 256, 0, stream>>>(statics, xy, BN);
        edge_det_compact<<<egrid, 256, 0, stream>>>(ei, xy, det, E, N);
    } else {
        edge_det_direct<<<egrid, 256, 0, stream>>>(ei, statics, det, E, N);
    }
}